// TaggingSupervisedCRFLossBuilder_67121748902287
// MI455X (gfx1250) — compile-verified
//
#include <hip/hip_runtime.h>

#define BB 64
#define TT 512
#define KK 64

typedef float v2f __attribute__((ext_vector_type(2)));
typedef float v8f __attribute__((ext_vector_type(8)));

// ---------------------------------------------------------------------------
// Kernel 1: gold-path score  ws[b] = em_score[b] + tr_score[b]
// ---------------------------------------------------------------------------
__global__ void crf_path_score(const float* __restrict__ tw,
                               const float* __restrict__ trans,
                               const int*   __restrict__ gold,
                               const int*   __restrict__ lengths,
                               float*       __restrict__ score) {
  const int b   = blockIdx.x;
  const int tid = threadIdx.x;
  const int len = lengths[b];
  const int*   g   = gold + b * TT;
  const float* twb = tw + (size_t)b * TT * KK;

  float local = 0.0f;
  for (int t = tid; t < TT; t += blockDim.x) {
    if (t < len) {
      int gt = g[t];
      local += twb[t * KK + gt];              // emission score
      if (t >= 1) {
        int gp = g[t - 1];
        local += trans[gp * KK + gt];         // transition score
      }
    }
  }
  __shared__ float red[128];
  red[tid] = local;
  __syncthreads();
  for (int s = 64; s > 0; s >>= 1) {
    if (tid < s) red[tid] += red[tid + s];
    __syncthreads();
  }
  if (tid == 0) score[b] = red[0];
}

// ---------------------------------------------------------------------------
// Kernel 2: forward recursion (log-partition) via f32 WMMA in exp-space.
// Grid: 4 blocks (16 batch rows each), 128 threads = 4 waves (one N-tile each).
// Alpha is carried offset-split:  true_alpha[m,n] = atil[m,n] + off[m],
// which keeps the per-step update shuffle-free in both layouts.
// ---------------------------------------------------------------------------
__global__ __launch_bounds__(128)
void crf_forward(const float* __restrict__ tw,
                 const float* __restrict__ trans,
                 const int*   __restrict__ lengths,
                 const float* __restrict__ score,
                 float*       __restrict__ out) {
  __shared__ float abuf[2][16 * KK];          // double-buffered atil tile (2 x 4KB)

  const int tid  = threadIdx.x;
  const int wave = tid >> 5;                  // 0..3: owns N columns [16w, 16w+16)
  const int lane = tid & 31;
  const int l16  = lane & 15;
  const int hi   = lane >> 4;                 // 0 = lanes 0-15, 1 = lanes 16-31
  const int base = blockIdx.x * 16;           // batch-row base of this block
  const int nb   = wave * 16;
  const int n    = nb + l16;                  // this lane's column (B/C/D layouts)

  // B fragments: E = exp(Tr), constant over all time steps, kept in VGPRs.
  // B 4x16 layout: VGPR0 = K{0|2} (low|high half), VGPR1 = K{1|3}, lanes = N.
  v2f e[16];
#pragma unroll
  for (int s = 0; s < 16; ++s) {
    const int kr = 4 * s + 2 * hi;
    e[s].x = __expf(trans[kr * KK + n]);
    e[s].y = __expf(trans[(kr + 1) * KK + n]);
  }

  // D-layout state: VGPR r holds row (r + 8*hi), column n.
  float        atil[8];                       // alpha minus per-row offset
  int          lenD[8];
  const float* rowp[8];
#pragma unroll
  for (int r = 0; r < 8; ++r) {
    const int brow = base + r + 8 * hi;
    lenD[r] = lengths[brow];
    rowp[r] = tw + (size_t)brow * TT * KK + n;
    atil[r] = rowp[r][0];                     // alpha0 = tag_weights[:, 0, :]
  }

  // A-layout state: this lane tracks row l16's accumulated offset.
  float     off  = 0.0f;
  const int lenA = lengths[base + l16];

  int p = 0;
  for (int t = 1; t < TT; ++t) {
    // emission for this step (issue global loads early, overlap with LDS+barrier)
    float emit[8];
#pragma unroll
    for (int r = 0; r < 8; ++r)
      emit[r] = rowp[r][t * KK];

    // 1) publish atil tile (D layout -> row-major LDS)
#pragma unroll
    for (int r = 0; r < 8; ++r)
      abuf[p][(r + 8 * hi) * KK + n] = atil[r];
    __syncthreads();

    // 2) A fragments: lane l16 holds row l16, K-cols {4s+2hi, 4s+2hi+1}
    v2f af[16];
#pragma unroll
    for (int s = 0; s < 16; ++s) {
      const float* src = &abuf[p][l16 * KK + 4 * s + 2 * hi];
      af[s].x = src[0];
      af[s].y = src[1];
    }

    // 3) row max of row l16 (lane pair {l16, l16+16} covers the 64 values)
    float mr = af[0].x;
#pragma unroll
    for (int s = 0; s < 16; ++s) {
      mr = fmaxf(mr, af[s].x);
      mr = fmaxf(mr, af[s].y);
    }
    mr = fmaxf(mr, __shfl_xor(mr, 16, 32));

    // 4) S = exp(atil - mr) x E   via 16 chained f32 WMMAs (K consumed 4 at a time)
    v8f S = {0.f, 0.f, 0.f, 0.f, 0.f, 0.f, 0.f, 0.f};
#pragma unroll
    for (int s = 0; s < 16; ++s) {
      v2f pf;
      pf.x = __expf(af[s].x - mr);
      pf.y = __expf(af[s].y - mr);
      S = __builtin_amdgcn_wmma_f32_16x16x4_f32(
          /*neg_a=*/false, pf, /*neg_b=*/false, e[s],
          /*c_mod=*/(short)0, S, /*reuse_a=*/false, /*reuse_b=*/false);
    }

    // 5) offset-split update, masked by t < length, no cross-lane traffic:
    //    true alpha' = log(S) + mr + off + emit;  choose off' = off + mr
    //    => stored atil' = log(S) + emit.
    off = (t < lenA) ? (off + mr) : off;      // A-layout row offset
#pragma unroll
    for (int r = 0; r < 8; ++r) {
      const float na = __logf(S[r]) + emit[r];
      atil[r] = (t < lenD[r]) ? na : atil[r]; // D-layout stored value
    }
    p ^= 1;
  }

  // Final: log_partition[b] = off[b] + logsumexp_n atil[b, n]
#pragma unroll
  for (int r = 0; r < 8; ++r)
    abuf[0][(r + 8 * hi) * KK + n] = atil[r];
  __syncthreads();

  if (tid < 16) {                             // wave 0, lane tid: owns off for row tid
    const float* row = &abuf[0][tid * KK];
    float m = row[0];
    for (int j = 1; j < KK; ++j) m = fmaxf(m, row[j]);
    float ssum = 0.0f;
    for (int j = 0; j < KK; ++j) ssum += __expf(row[j] - m);
    const float lp = off + m + __logf(ssum);
    const int b = base + tid;
    out[b] = lp - score[b];
  }
}

// ---------------------------------------------------------------------------
extern "C" void kernel_launch(void* const* d_in, const int* in_sizes, int n_in,
                              void* d_out, int out_size, void* d_ws, size_t ws_size,
                              hipStream_t stream) {
  const float* tw      = (const float*)d_in[0];   // (64, 512, 64) f32
  const float* trans   = (const float*)d_in[1];   // (64, 64) f32
  const int*   gold    = (const int*)d_in[2];     // (64, 512) int
  const int*   lengths = (const int*)d_in[3];     // (64,) int
  float* out = (float*)d_out;                     // (64,) f32
  float* ws  = (float*)d_ws;                      // 64 floats scratch

  crf_path_score<<<BB, 128, 0, stream>>>(tw, trans, gold, lengths, ws);
  crf_forward<<<4, 128, 0, stream>>>(tw, trans, lengths, ws, out);
}